// ChiSq_34789235098204
// MI455X (gfx1250) — compile-verified
//
#include <hip/hip_runtime.h>
#include <stdint.h>

// ---------------- problem constants (from reference) ----------------
#define NUM_BINS 16
#define NT       17           // NUM_BINS + 1 targets/edges
#define RROWS    256          // B*C = 128*2 independent rows
#define FLEN     65537        // F = 32*4096/2 + 1
#define FOURDF   0.125f       // 4 * (1/FFTLENGTH) = 4/32

// ---------------- tiling ----------------
#define THREADS  256          // 8 waves (wave32)
#define VE       8            // elements per thread
#define TILE     (THREADS * VE)              // 2048 elements per chunk
#define NCH      ((FLEN + TILE - 1) / TILE)  // 33 chunks per row

// ---------------- CDNA5 TDM (tensor_load_to_lds) ----------------
#if defined(__HIP_DEVICE_COMPILE__) && __has_builtin(__builtin_amdgcn_tensor_load_to_lds)
#define HAVE_TDM 1
#else
#define HAVE_TDM 0
#endif

#if HAVE_TDM
typedef unsigned int tdm_u32x4 __attribute__((ext_vector_type(4)));
typedef int          tdm_i32x4 __attribute__((ext_vector_type(4)));
typedef int          tdm_i32x8 __attribute__((ext_vector_type(8)));

// 1-D (degenerate 2-D) DMA of n fp32 elements: global -> LDS.
// D# layout per CDNA5 ISA ch.8: group0 = {flags, lds_addr, global_addr, type=2},
// group1 = {data_size=4B, tensor_dim0=n, tensor_dim1=1, tile_dim0=n, tile_dim1=1}.
static __device__ __forceinline__ void tdm_load_row(const float* gsrc, float* ldst, int n) {
  unsigned long long ga = (unsigned long long)(uintptr_t)gsrc;
  unsigned int       la = (unsigned int)(uintptr_t)ldst;   // flat LDS addr: low 32 bits = LDS offset
  tdm_u32x4 g0;
  g0[0] = 1u;                                              // count=1 (valid), no gather
  g0[1] = la;                                              // lds_addr
  g0[2] = (unsigned int)(ga & 0xFFFFFFFFull);              // global_addr[31:0]
  g0[3] = ((unsigned int)(ga >> 32) & 0x01FFFFFFu) | 0x80000000u; // addr[56:32] | type=2
  tdm_i32x8 g1;
  g1[0] = 0x00020000;                                      // data_size=2 (4 bytes), no pad/iter/mask
  g1[1] = (int)(((unsigned)n & 0xFFFFu) << 16);            // tensor_dim0[15:0] @ bits 63:48
  g1[2] = (int)((((unsigned)n >> 16) & 0xFFFFu) | (1u << 16)); // tensor_dim0 hi | tensor_dim1=1
  g1[3] = (int)(((unsigned)n & 0xFFFFu) << 16);            // tensor_dim1 hi=0 | tile_dim0=n
  g1[4] = 1;                                               // tile_dim1=1, tile_dim2 unused
  g1[5] = n;                                               // tensor_dim0_stride (one row; benign)
  g1[6] = 0;
  g1[7] = 0;
  tdm_i32x4 z4 = {0, 0, 0, 0};
#if __clang_major__ >= 23
  tdm_i32x8 z8 = {0, 0, 0, 0, 0, 0, 0, 0};
  __builtin_amdgcn_tensor_load_to_lds(g0, g1, z4, z4, z8, 0);
#else
  __builtin_amdgcn_tensor_load_to_lds(g0, g1, z4, z4, 0);
#endif
}
#endif // HAVE_TDM

// Stage one chunk (n valid elements) of all five streams into LDS.
static __device__ __forceinline__ void load_tile(
    const float* gh, const float* gi, const float* gs, const float* gj, const float* gp,
    float* lh, float* li, float* ls, float* lj, float* lp, int n) {
#if HAVE_TDM
  if (threadIdx.x == 0) {               // TDM ignores EXEC; one issue per workgroup
    tdm_load_row(gh, lh, n);
    tdm_load_row(gi, li, n);
    tdm_load_row(gs, ls, n);
    tdm_load_row(gj, lj, n);
    tdm_load_row(gp, lp, n);
  }
  __builtin_amdgcn_s_wait_tensorcnt(0); // issuer drains TENSORcnt; others pass (cnt==0)
  __syncthreads();                      // publish LDS to all waves
#else
  for (int i = (int)threadIdx.x; i < TILE; i += THREADS) {
    lh[i] = (i < n) ? gh[i] : 0.0f;
    li[i] = (i < n) ? gi[i] : 0.0f;
    ls[i] = (i < n) ? gs[i] : 0.0f;
    lj[i] = (i < n) ? gj[i] : 0.0f;
    lp[i] = (i < n) ? gp[i] : 1.0f;
  }
  __syncthreads();
#endif
}

// Per-element weights. |h*conj(h)| = hre^2+him^2 (imag cancels exactly);
// |q*conj(s)| = |h||s|/sqrt(Th) -> keep the raw (unnormalized) q-weight here.
static __device__ __forceinline__ void calc_w(float ar, float ai, float br, float bi,
                                              float p, float& wh, float& wq) {
  float ip = 1.0f / p;
  wh = FOURDF * (ar * ar + ai * ai) * ip;
  float re = ar * br + ai * bi;
  float im = ai * br - ar * bi;
  wq = FOURDF * sqrtf(re * re + im * im) * ip;
}

// Inclusive Hillis-Steele scan of per-thread (h,q) partials in LDS.
// Returns the thread's exclusive prefix and the chunk totals.
static __device__ __forceinline__ void block_scan2(float oh, float oq, float* sh, float* sq,
                                                   float& eh, float& eq, float& th, float& tq) {
  const int tid = (int)threadIdx.x;
  sh[tid] = oh; sq[tid] = oq;
  __syncthreads();
  for (int off = 1; off < THREADS; off <<= 1) {
    float ah = 0.0f, aq = 0.0f;
    if (tid >= off) { ah = sh[tid - off]; aq = sq[tid - off]; }
    __syncthreads();
    if (tid >= off) { sh[tid] += ah; sq[tid] += aq; }
    __syncthreads();
  }
  eh = tid ? sh[tid - 1] : 0.0f;
  eq = tid ? sq[tid - 1] : 0.0f;
  th = sh[THREADS - 1];
  tq = sq[THREADS - 1];
}

// ---------------- Pass A: per-chunk partial sums ----------------
__global__ __launch_bounds__(THREADS)
void chisq_pass_partials(const float* __restrict__ hre, const float* __restrict__ him,
                         const float* __restrict__ sre, const float* __restrict__ sim,
                         const float* __restrict__ psd,
                         float* __restrict__ chunk_h, float* __restrict__ chunk_q) {
  __shared__ float lh[TILE], li[TILE], ls[TILE], lj[TILE], lp[TILE];
  __shared__ float sh[THREADS], sq[THREADS];
  const int r  = (int)blockIdx.y;
  const int ch = (int)blockIdx.x;
  const long long rb = (long long)r * FLEN;
  const int tb = ch * TILE;
  int n = FLEN - tb; if (n > TILE) n = TILE;

  load_tile(hre + rb + tb, him + rb + tb, sre + rb + tb, sim + rb + tb, psd + rb + tb,
            lh, li, ls, lj, lp, n);

  const int i0 = (int)threadIdx.x * VE;
  float runh = 0.0f, runq = 0.0f;
  #pragma unroll
  for (int j = 0; j < VE; ++j) {
    int idx = i0 + j;
    if (idx < n) {
      float wh, wq;
      calc_w(lh[idx], li[idx], ls[idx], lj[idx], lp[idx], wh, wq);
      runh += wh; runq += wq;
    }
  }
  float eh, eq, th, tq;
  block_scan2(runh, runq, sh, sq, eh, eq, th, tq);
  if (threadIdx.x == 0) {
    chunk_h[r * NCH + ch] = th;
    chunk_q[r * NCH + ch] = tq;
  }
}

// ---------------- Pass B: per-row chunk prefix scan + totals + edge defaults ----------------
__global__ __launch_bounds__(RROWS)
void chisq_pass_prefix(const float* __restrict__ chunk_h, const float* __restrict__ chunk_q,
                       float* __restrict__ pref_h, float* __restrict__ pref_q,
                       float* __restrict__ tot, float* __restrict__ vbuf) {
  const int r = (int)threadIdx.x;
  float p = 0.0f, q = 0.0f;
  for (int c = 0; c < NCH; ++c) {
    pref_h[r * NCH + c] = p; p += chunk_h[r * NCH + c];
    pref_q[r * NCH + c] = q; q += chunk_q[r * NCH + c];
  }
  float Th   = p;
  float invs = 1.0f / sqrtf(Th);
  float Tq   = q * invs;                 // si_q[-1] after q-normalization
  tot[2 * r]     = Th;
  tot[2 * r + 1] = Tq;
  for (int k = 0; k < NT; ++k)           // default = clipped edge (edges==F-1) value
    vbuf[r * NT + k] = Tq;
}

// ---------------- Pass C: full scan + fused searchsorted/gather ----------------
__global__ __launch_bounds__(THREADS)
void chisq_pass_scan(const float* __restrict__ hre, const float* __restrict__ him,
                     const float* __restrict__ sre, const float* __restrict__ sim,
                     const float* __restrict__ psd,
                     const float* __restrict__ pref_h, const float* __restrict__ pref_q,
                     const float* __restrict__ tot, float* __restrict__ vbuf) {
  __shared__ float lh[TILE], li[TILE], ls[TILE], lj[TILE], lp[TILE];
  __shared__ float sh[THREADS], sq[THREADS];
  const int r  = (int)blockIdx.y;
  const int ch = (int)blockIdx.x;
  const long long rb = (long long)r * FLEN;
  const int tb = ch * TILE;
  int n = FLEN - tb; if (n > TILE) n = TILE;

  load_tile(hre + rb + tb, him + rb + tb, sre + rb + tb, sim + rb + tb, psd + rb + tb,
            lh, li, ls, lj, lp, n);

  const int i0 = (int)threadIdx.x * VE;
  float rh[VE], rq[VE];
  float runh = 0.0f, runq = 0.0f;
  #pragma unroll
  for (int j = 0; j < VE; ++j) {
    int idx = i0 + j;
    float wh = 0.0f, wq = 0.0f;
    if (idx < n) calc_w(lh[idx], li[idx], ls[idx], lj[idx], lp[idx], wh, wq);
    runh += wh; rh[j] = runh;
    runq += wq; rq[j] = runq;
  }
  float eh, eq, th, tq;
  block_scan2(runh, runq, sh, sq, eh, eq, th, tq);

  const float Th   = tot[2 * r];
  const float invs = 1.0f / sqrtf(Th);
  const float s16  = (float)NUM_BINS / Th;
  const float Bh   = pref_h[r * NCH + ch] + eh;   // exclusive si_h at thread start
  const float Bq   = pref_q[r * NCH + ch] + eq;   // exclusive raw si_q at thread start
  float* vout = vbuf + r * NT;

  float prev = 0.0f;
  #pragma unroll
  for (int j = 0; j < VE; ++j) {
    int idx = i0 + j;
    if (idx >= n) break;
    float l = Bh + prev;        // exclusive cumsum (element idx-1 inclusive)
    float u = Bh + rh[j];       // inclusive cumsum at element idx
    prev = rh[j];
    if (u > l) {
      // element idx is edge e_k for every target t_k in [l, u); window via arithmetic + exact test
      int k0 = (int)floorf(l * s16) - 1; if (k0 < 0) k0 = 0;
      int k1 = (int)floorf(u * s16) + 1; if (k1 > NUM_BINS) k1 = NUM_BINS;
      float iq = (Bq + rq[j]) * invs;   // si_q[e_k]
      for (int k = k0; k <= k1; ++k) {
        float t = Th * ((float)k * (1.0f / NUM_BINS));
        if (l <= t && t < u) vout[k] = iq;   // exactly one writer per k
      }
    }
  }
}

// ---------------- Pass D: chi-square + outputs ----------------
__global__ __launch_bounds__(RROWS)
void chisq_pass_final(const float* __restrict__ vbuf, const float* __restrict__ tot,
                      float* __restrict__ out) {
  const int r = (int)threadIdx.x;
  const float Tq   = tot[2 * r + 1];
  const float mean = Tq * (1.0f / NUM_BINS);
  float acc = 0.0f;
  for (int k = 0; k < NUM_BINS; ++k) {
    float d = vbuf[r * NT + k + 1] - vbuf[r * NT + k] - mean;
    acc += d * d;
  }
  out[r]         = Tq;                                        // total
  out[RROWS + r] = (float)NUM_BINS * acc / (float)(NUM_BINS - 1); // chisq
}

// ---------------- host launcher ----------------
extern "C" void kernel_launch(void* const* d_in, const int* in_sizes, int n_in,
                              void* d_out, int out_size, void* d_ws, size_t ws_size,
                              hipStream_t stream) {
  (void)in_sizes; (void)n_in; (void)out_size; (void)ws_size;
  const float* hre = (const float*)d_in[0];
  const float* him = (const float*)d_in[1];
  const float* sre = (const float*)d_in[2];
  const float* sim = (const float*)d_in[3];
  const float* psd = (const float*)d_in[4];
  float* out = (float*)d_out;

  float* ws      = (float*)d_ws;
  float* chunk_h = ws;                          // [RROWS][NCH]
  float* chunk_q = chunk_h + RROWS * NCH;
  float* pref_h  = chunk_q + RROWS * NCH;
  float* pref_q  = pref_h + RROWS * NCH;
  float* tot     = pref_q + RROWS * NCH;        // [RROWS][2] = {total_h, total_q}
  float* vbuf    = tot + 2 * RROWS;             // [RROWS][NT] edge si_q values

  dim3 grid(NCH, RROWS);
  chisq_pass_partials<<<grid, THREADS, 0, stream>>>(hre, him, sre, sim, psd, chunk_h, chunk_q);
  chisq_pass_prefix<<<1, RROWS, 0, stream>>>(chunk_h, chunk_q, pref_h, pref_q, tot, vbuf);
  chisq_pass_scan<<<grid, THREADS, 0, stream>>>(hre, him, sre, sim, psd, pref_h, pref_q, tot, vbuf);
  chisq_pass_final<<<1, RROWS, 0, stream>>>(vbuf, tot, out);
}